// my_regression_low_68788196212844
// MI455X (gfx1250) — compile-verified
//
#include <hip/hip_runtime.h>
#include <hip/hip_bf16.h>

typedef __attribute__((ext_vector_type(2))) float v2f;
typedef __attribute__((ext_vector_type(8))) float v8f;

#define LEAD 8
#define DDIM 64800
#define KDIM 16
#define BATCH 256
#define NSPLIT 54              // d-axis splits per batch row-tile
#define CHUNK (DDIM / NSPLIT)  // 1200, multiple of 4

// One wave (32 threads) per (n, d-split). Computes the 16x16 partial
// Y_tile = X_tile(16 x CHUNK) * C_chunk(CHUNK x 16) with fp32 WMMA
// (V_WMMA_F32_16X16X4_F32), folds it with W[m,k] = A[l,k]*B[c,k]
// (m = l*2+c), wave-reduces, writes one scalar partial.
//
// X (1.06 GB) is streamed exactly once -> non-temporal loads so the
// 192 MB L2 stays dedicated to C (4.1 MB, reused by all 256*54 waves).
__global__ __launch_bounds__(32) void cp_core_partial_kernel(
    const float* __restrict__ X,   // [BATCH*16][DDIM] row-major
    const float* __restrict__ A,   // [LEAD][KDIM]
    const float* __restrict__ B,   // [2][KDIM]
    const float* __restrict__ C,   // [DDIM][KDIM]
    float* __restrict__ partial)   // [BATCH][NSPLIT]
{
    const int n    = blockIdx.x;          // batch index == 16-row tile
    const int s    = blockIdx.y;          // d-split
    const int lane = threadIdx.x;         // 0..31 (wave32)

    const int m    = lane & 15;           // A-frag row / B-frag column
    const int koff = (lane >> 4) << 1;    // 0 for lanes 0-15, 2 for 16-31

    const int d0 = s * CHUNK;

    // A-frag source: X row (n*16 + m), consecutive d pair starting at d+koff
    const float* xp = X + ((size_t)n * 16 + m) * DDIM + d0 + koff;
    // B-frag source: C rows (d+koff, d+koff+1) at column m
    const float* cp = C + (size_t)(d0 + koff) * KDIM + m;

    v8f acc = {0.f, 0.f, 0.f, 0.f, 0.f, 0.f, 0.f, 0.f};

#pragma unroll 8
    for (int it = 0; it < CHUNK / 4; ++it) {
        // streaming operand: non-temporal (TH=NT) global_load_b64
        v2f afrag = __builtin_nontemporal_load((const v2f*)xp);
        v2f bfrag;
        bfrag.x = cp[0];                         // C[(d+koff)  ][m]  (L2-resident)
        bfrag.y = cp[KDIM];                      // C[(d+koff+1)][m]
        acc = __builtin_amdgcn_wmma_f32_16x16x4_f32(
            /*neg_a=*/false, afrag, /*neg_b=*/false, bfrag,
            /*c_mod=*/(short)0, acc, /*reuse_a=*/false, /*reuse_b=*/false);
        xp += 4;                // advance 4 d-values
        cp += 4 * KDIM;
    }

    // Fold 16x16 partial with W[mrow,k] = A[l,k]*B[c,k], mrow = l*2+c.
    // C/D layout: lane holds column k = lane&15; VGPR r holds row
    // mrow = r + (lane>=16 ? 8 : 0).
    const int k    = lane & 15;
    const int half = (lane >> 4) << 3;
    float sum = 0.f;
#pragma unroll
    for (int r = 0; r < 8; ++r) {
        const int mrow = half + r;
        const float w = A[(mrow >> 1) * KDIM + k] * B[(mrow & 1) * KDIM + k];
        sum += acc[r] * w;
    }

    // wave32 butterfly reduction
#pragma unroll
    for (int off = 16; off >= 1; off >>= 1)
        sum += __shfl_xor(sum, off, 32);

    if (lane == 0)
        partial[(size_t)n * NSPLIT + s] = sum;
}

// Deterministic fixed-order reduction of the NSPLIT partials per n, plus b.
__global__ __launch_bounds__(BATCH) void cp_core_finalize_kernel(
    const float* __restrict__ partial,  // [BATCH][NSPLIT]
    const float* __restrict__ bscalar,  // [1]
    float* __restrict__ out)            // [BATCH]
{
    const int n = threadIdx.x;
    float s = 0.f;
#pragma unroll 1
    for (int i = 0; i < NSPLIT; ++i)
        s += partial[(size_t)n * NSPLIT + i];
    out[n] = s + bscalar[0];
}

extern "C" void kernel_launch(void* const* d_in, const int* in_sizes, int n_in,
                              void* d_out, int out_size, void* d_ws, size_t ws_size,
                              hipStream_t stream) {
    const float* X = (const float*)d_in[0];   // [256][8][2][64800]
    const float* A = (const float*)d_in[1];   // [8][16]
    const float* B = (const float*)d_in[2];   // [2][16]
    const float* C = (const float*)d_in[3];   // [64800][16]
    const float* b = (const float*)d_in[4];   // [1]
    float* out     = (float*)d_out;           // [256]
    float* partial = (float*)d_ws;            // BATCH*NSPLIT floats = 55 KB

    dim3 grid1(BATCH, NSPLIT);
    cp_core_partial_kernel<<<grid1, dim3(32), 0, stream>>>(X, A, B, C, partial);
    cp_core_finalize_kernel<<<1, dim3(BATCH), 0, stream>>>(partial, b, out);
}